// Net_2370821948028
// MI455X (gfx1250) — compile-verified
//
#include <hip/hip_runtime.h>
#include <hip/hip_bf16.h>
#include <math.h>

// ---------------- problem constants (match reference) ----------------
#define G_    32
#define NPG_  1024
#define KC_   10          // pooling clusters K
#define H_    10          // hidden width
#define N_    32768       // G*NPG nodes
#define E_    131072      // N*DEG edges
#define EPG_  4096        // edges per graph (block-structured edge list)
#define BNEPS 1e-5f

// WMMA f32 fragment types: A/B 16x4 / 4x16 f32 = 2 VGPRs, C/D 16x16 f32 = 8 VGPRs
typedef __attribute__((ext_vector_type(2))) float v2f;
typedef __attribute__((ext_vector_type(8))) float v8f;

// =====================================================================
// Tiled linear: out[M,DOUT] = in[M,DIN] @ w[DIN,DOUT], DIN/DOUT <= 16
// compile-time. One wave per 16-row tile, K in chunks of 4 via
// V_WMMA_F32_16X16X4_F32 (native fp32: workload is latency-bound, so no
// precision downgrade). Guards are clamp-index + select (v_cndmask),
// never exec-masked branches; K loop fully unrolls with imm offsets.
// =====================================================================
template <int DIN, int DOUT>
__global__ __launch_bounds__(256)
void k_wmma_linear(const float* __restrict__ in, const float* __restrict__ w,
                   float* __restrict__ out, int M) {
  int wave = (int)((blockIdx.x * blockDim.x + threadIdx.x) >> 5);
  int lane = threadIdx.x & 31;
  int row0 = wave * 16;
  if (row0 >= M) return;                 // wave-uniform: EXEC all-ones at WMMA
  int lid = lane & 15;                   // A row (m) and B/C/D column (n)
  int hi  = lane >> 4;                   // half-wave select
  bool nok = (lid < DOUT);
  int lcl  = nok ? lid : 0;              // clamped column for B loads
  const float* arow = in + (size_t)(row0 + lid) * DIN;  // per-lane A row base
  const float* wcol = w + lcl;                          // per-lane B col base
  v8f acc = {};
  constexpr int NK = (DIN + 3) >> 2;
#pragma unroll
  for (int kc = 0; kc < NK; ++kc) {
    v2f a, b;
#pragma unroll
    for (int j = 0; j < 2; ++j) {
      int k = kc * 4 + hi * 2 + j;       // ISA A-layout: VGPRj -> K=j / K=j+2 per half-wave
      bool kok = (k < DIN);
      int kcl = kok ? k : 0;             // clamped: load always in-bounds
      float av = arow[kcl];
      float bv = wcol[(size_t)kcl * DOUT];
      a[j] = kok ? av : 0.0f;
      b[j] = (kok && nok) ? bv : 0.0f;
    }
    acc = __builtin_amdgcn_wmma_f32_16x16x4_f32(false, a, false, b, (short)0, acc,
                                                false, false);
  }
  if (nok) {
    float* orow = out + (size_t)(row0 + hi * 8) * DOUT + lid;
#pragma unroll
    for (int j = 0; j < 8; ++j)          // C/D: VGPRj -> row j / j+8 per half-wave
      orow[(size_t)j * DOUT] = acc[j];
  }
}

// =====================================================================
// p1_x[g,k,d] = sum_n s[g,n,k] * x13[g,n,d]  -- per-graph s^T x,
// 16x16 output, K=1024 via 256 WMMA accumulation steps. Wave per graph.
// Hot loop: 4 unconditional loads + 1 v_wmma (masks hoisted).
// =====================================================================
__global__ __launch_bounds__(256)
void k_p1x_wmma(const float* __restrict__ s, const float* __restrict__ x,
                float* __restrict__ p1x) {
  int wave = (int)((blockIdx.x * blockDim.x + threadIdx.x) >> 5);
  if (wave >= G_) return;
  int lane = threadIdx.x & 31;
  int lid = lane & 15, hi = lane >> 4;
  bool ok = (lid < KC_);                 // KC_ == H_ == 10
  int lcl = ok ? lid : 0;
  const float* sg = s + (size_t)wave * NPG_ * KC_ + lcl;  // per-lane col base
  const float* xg = x + (size_t)wave * NPG_ * H_  + lcl;
  v8f acc = {};
#pragma unroll 4
  for (int nc = 0; nc < NPG_ / 4; ++nc) {
    v2f a, b;
#pragma unroll
    for (int j = 0; j < 2; ++j) {
      int node = nc * 4 + hi * 2 + j;
      float av = sg[(size_t)node * KC_];  // A[k=lid][node]
      float bv = xg[(size_t)node * H_];   // B[node][d=lid]
      a[j] = ok ? av : 0.0f;
      b[j] = ok ? bv : 0.0f;
    }
    acc = __builtin_amdgcn_wmma_f32_16x16x4_f32(false, a, false, b, (short)0, acc,
                                                false, false);
  }
  if (ok) {
#pragma unroll
    for (int j = 0; j < 8; ++j) {
      int k = hi * 8 + j;
      if (k < KC_) p1x[((size_t)wave * KC_ + k) * H_ + lid] = acc[j];
    }
  }
}

// =====================================================================
// Scatter degree + 2m reduction
// =====================================================================
__global__ void k_degree(const int* __restrict__ src, const int* __restrict__ dst,
                         const float* __restrict__ ea, float* __restrict__ deg,
                         float* __restrict__ two_m) {
  __shared__ float ls;
  if (threadIdx.x == 0) ls = 0.0f;
  __syncthreads();
  int e = blockIdx.x * blockDim.x + threadIdx.x;
  float w = 0.0f;
  if (e < E_) {
    w = ea[e];
    atomicAdd(&deg[dst[e]], w);
  }
  (void)src;
  for (int o = 16; o > 0; o >>= 1) w += __shfl_down(w, o, 32);
  if ((threadIdx.x & 31) == 0) atomicAdd(&ls, w);
  __syncthreads();
  if (threadIdx.x == 0) atomicAdd(two_m, ls);
}

__global__ void k_dis(const float* __restrict__ deg, float* __restrict__ dis) {
  int n = blockIdx.x * blockDim.x + threadIdx.x;
  if (n < N_) dis[n] = rsqrtf(deg[n] + 1.0f);   // self-loop added; deg[] stays = deg_w
}

__global__ void k_zero(float* __restrict__ p, int n) {
  int i = blockIdx.x * blockDim.x + threadIdx.x;
  if (i < n) p[i] = 0.0f;
}

// out[dst] += h[src] * (dis[src]*dis[dst]*ea)   (10 channels per edge)
__global__ void k_gcn_scatter(const int* __restrict__ src, const int* __restrict__ dst,
                              const float* __restrict__ ea, const float* __restrict__ dis,
                              const float* __restrict__ h, float* __restrict__ out) {
  int e = blockIdx.x * blockDim.x + threadIdx.x;
  if (e >= E_) return;
  int s = src[e], d = dst[e];
  const float* hr = h + (size_t)s * H_;
  __builtin_prefetch(hr, 0, 1);                 // global_prefetch_b8
  float coef = dis[s] * dis[d] * ea[e];
  float* orow = out + (size_t)d * H_;
#pragma unroll
  for (int c = 0; c < H_; ++c) atomicAdd(&orow[c], hr[c] * coef);
}

// y += h*dis^2 + bias, then accumulate per-channel sum/sumsq (wave-reduced)
__global__ void k_self_stats(const float* __restrict__ h, const float* __restrict__ dis,
                             const float* __restrict__ bias, float* __restrict__ y,
                             float* __restrict__ stats) {
  __shared__ float ls[2 * H_];
  if (threadIdx.x < 2 * H_) ls[threadIdx.x] = 0.0f;
  __syncthreads();
  int n = blockIdx.x * blockDim.x + threadIdx.x;
  int lane = threadIdx.x & 31;
  bool act = (n < N_);
  float d2 = act ? dis[n] * dis[n] : 0.0f;
  float v[H_];
#pragma unroll
  for (int c = 0; c < H_; ++c) {
    float t = act ? (y[(size_t)n * H_ + c] + h[(size_t)n * H_ + c] * d2 + bias[c]) : 0.0f;
    if (act) y[(size_t)n * H_ + c] = t;
    v[c] = t;
  }
#pragma unroll
  for (int c = 0; c < H_; ++c) {
    float s = v[c], q = v[c] * v[c];
    for (int o = 16; o > 0; o >>= 1) { s += __shfl_down(s, o, 32); q += __shfl_down(q, o, 32); }
    if (lane == 0) { atomicAdd(&ls[c], s); atomicAdd(&ls[H_ + c], q); }
  }
  __syncthreads();
  if (threadIdx.x < H_) {
    atomicAdd(&stats[threadIdx.x],      ls[threadIdx.x]);
    atomicAdd(&stats[16 + threadIdx.x], ls[H_ + threadIdx.x]);
  }
}

// BatchNorm apply using accumulated sum/sumsq (biased variance, as jnp.var)
__global__ void k_bn_apply(const float* __restrict__ y, const float* __restrict__ stats,
                           const float* __restrict__ gam, const float* __restrict__ bet,
                           float* __restrict__ xout, int M) {
  int i = blockIdx.x * blockDim.x + threadIdx.x;
  if (i >= M * H_) return;
  int c = i % H_;
  float inv = 1.0f / (float)M;
  float mean = stats[c] * inv;
  float var  = stats[16 + c] * inv - mean * mean;
  xout[i] = (y[i] - mean) * rsqrtf(var + BNEPS) * gam[c] + bet[c];
}

// per-graph max over 1024 nodes of concat(x11,x12,x13) -> [G,30]
__global__ void k_graph_max(const float* __restrict__ a, const float* __restrict__ b,
                            const float* __restrict__ c3, float* __restrict__ out) {
  __shared__ float red[8][32];
  int g = blockIdx.x;
  int c = threadIdx.x & 31;
  int ch = threadIdx.x >> 5;
  float m = -3.4e38f;
  const float* p = (c < 10) ? a : (c < 20) ? b : c3;
  int cc = c % 10;
  if (c < 30) {
    for (int n = ch * 128; n < (ch + 1) * 128; ++n)
      m = fmaxf(m, p[(size_t)(g * NPG_ + n) * H_ + cc]);
  }
  red[ch][c] = m;
  __syncthreads();
  if (ch == 0 && c < 30) {
    float mm = red[0][c];
#pragma unroll
    for (int k = 1; k < 8; ++k) mm = fmaxf(mm, red[k][c]);
    out[g * 30 + c] = mm;
  }
}

// row softmax over KC_ (bias folded in)
__global__ void k_softmax(float* __restrict__ z, const float* __restrict__ bias, int M) {
  int n = blockIdx.x * blockDim.x + threadIdx.x;
  if (n >= M) return;
  float v[KC_]; float mx = -3.4e38f;
#pragma unroll
  for (int c = 0; c < KC_; ++c) { v[c] = z[(size_t)n * KC_ + c] + bias[c]; mx = fmaxf(mx, v[c]); }
  float sm = 0.0f;
#pragma unroll
  for (int c = 0; c < KC_; ++c) { v[c] = expf(v[c] - mx); sm += v[c]; }
  float inv = 1.0f / sm;
#pragma unroll
  for (int c = 0; c < KC_; ++c) z[(size_t)n * KC_ + c] = v[c] * inv;
}

// sum_e dot(s[src],s[dst]) * ea  (edge part of modularity)
__global__ void k_mod_edge(const int* __restrict__ src, const int* __restrict__ dst,
                           const float* __restrict__ ea, const float* __restrict__ s,
                           float* __restrict__ qacc) {
  __shared__ float ls;
  if (threadIdx.x == 0) ls = 0.0f;
  __syncthreads();
  int e = blockIdx.x * blockDim.x + threadIdx.x;
  float q = 0.0f;
  if (e < E_) {
    const float* a = s + (size_t)src[e] * KC_;
    const float* b = s + (size_t)dst[e] * KC_;
    float d = 0.0f;
#pragma unroll
    for (int c = 0; c < KC_; ++c) d += a[c] * b[c];
    q = d * ea[e];
  }
  for (int o = 16; o > 0; o >>= 1) q += __shfl_down(q, o, 32);
  if ((threadIdx.x & 31) == 0) atomicAdd(&ls, q);
  __syncthreads();
  if (threadIdx.x == 0) atomicAdd(qacc, ls);
}

// dvec[k] = sum_n deg_w[n] * s[n,k]
__global__ void k_dvec(const float* __restrict__ s, const float* __restrict__ degw,
                       float* __restrict__ dvec) {
  __shared__ float ls[KC_];
  if (threadIdx.x < KC_) ls[threadIdx.x] = 0.0f;
  __syncthreads();
  int n = blockIdx.x * blockDim.x + threadIdx.x;
  float w = (n < N_) ? degw[n] : 0.0f;
#pragma unroll
  for (int c = 0; c < KC_; ++c) {
    float t = (n < N_) ? w * s[(size_t)n * KC_ + c] : 0.0f;
    for (int o = 16; o > 0; o >>= 1) t += __shfl_down(t, o, 32);
    if ((threadIdx.x & 31) == 0) atomicAdd(&ls[c], t);
  }
  __syncthreads();
  if (threadIdx.x < KC_) atomicAdd(&dvec[threadIdx.x], ls[threadIdx.x]);
}

// ml = -(q_edge/2m - sum_k (dvec[k]/2m)^2)
__global__ void k_ml(float* __restrict__ scal, const float* __restrict__ dvec) {
  if (threadIdx.x == 0 && blockIdx.x == 0) {
    float tm = scal[0];
    float q = scal[1] / tm;
    float t = 0.0f;
    for (int k = 0; k < KC_; ++k) { float r = dvec[k] / tm; t += r * r; }
    scal[3] = -(q - t);
  }
}

// sum_n sum_k -s*log(s+1e-15)
__global__ void k_entropy(const float* __restrict__ s, float* __restrict__ acc) {
  __shared__ float ls;
  if (threadIdx.x == 0) ls = 0.0f;
  __syncthreads();
  int n = blockIdx.x * blockDim.x + threadIdx.x;
  float e = 0.0f;
  if (n < N_) {
#pragma unroll
    for (int c = 0; c < KC_; ++c) { float p = s[(size_t)n * KC_ + c]; e -= p * logf(p + 1e-15f); }
  }
  for (int o = 16; o > 0; o >>= 1) e += __shfl_down(e, o, 32);
  if ((threadIdx.x & 31) == 0) atomicAdd(&ls, e);
  __syncthreads();
  if (threadIdx.x == 0) atomicAdd(acc, ls);
}

// p1_adj[g,k,l] = sum_{edges of g} s[src,k]*s[dst,l]*ea  -- edge list is
// block-structured so one block per graph, 10x10 LDS accumulator (ds_add_f32).
// Avoids materializing the 134 MB dense [G,1024,1024] adjacency.
__global__ void k_p1adj(const int* __restrict__ src, const int* __restrict__ dst,
                        const float* __restrict__ ea, const float* __restrict__ s,
                        float* __restrict__ p1adj) {
  __shared__ float acc[KC_ * KC_];
  int g = blockIdx.x, t = threadIdx.x;
  for (int i = t; i < KC_ * KC_; i += blockDim.x) acc[i] = 0.0f;
  __syncthreads();
  for (int e = g * EPG_ + t; e < (g + 1) * EPG_; e += blockDim.x) {
    float w = ea[e];
    const float* sa = s + (size_t)src[e] * KC_;
    const float* sb = s + (size_t)dst[e] * KC_;
    float av[KC_], bv[KC_];
#pragma unroll
    for (int c = 0; c < KC_; ++c) { av[c] = sa[c]; bv[c] = sb[c] * w; }
#pragma unroll
    for (int k = 0; k < KC_; ++k)
#pragma unroll
      for (int l = 0; l < KC_; ++l)
        atomicAdd(&acc[k * KC_ + l], av[k] * bv[l]);
  }
  __syncthreads();
  for (int i = t; i < KC_ * KC_; i += blockDim.x)
    p1adj[(size_t)g * KC_ * KC_ + i] = acc[i];
}

// anorm = D^-1/2 (A+I) D^-1/2 per graph
__global__ void k_anorm(const float* __restrict__ p1adj, float* __restrict__ anorm) {
  __shared__ float ah[KC_ * KC_];
  __shared__ float d2[KC_];
  int g = blockIdx.x, t = threadIdx.x;
  if (t < KC_ * KC_) { int i = t / KC_, j = t % KC_; ah[t] = p1adj[g * KC_ * KC_ + t] + (i == j ? 1.0f : 0.0f); }
  __syncthreads();
  if (t < KC_) { float s = 0.0f; for (int j = 0; j < KC_; ++j) s += ah[t * KC_ + j]; d2[t] = rsqrtf(s); }
  __syncthreads();
  if (t < KC_ * KC_) { int i = t / KC_, j = t % KC_; anorm[g * KC_ * KC_ + t] = ah[t] * d2[i] * d2[j]; }
}

// y[g] = anorm[g] @ hw[g] + bias  (10x10 x 10x10 per graph)
__global__ void k_gcn2(const float* __restrict__ anorm, const float* __restrict__ hw,
                       const float* __restrict__ bias, float* __restrict__ y) {
  int g = blockIdx.x, t = threadIdx.x;
  if (t >= KC_ * H_) return;
  int i = t / H_, d = t % H_;
  float a = 0.0f;
#pragma unroll
  for (int j = 0; j < KC_; ++j)
    a += anorm[(size_t)g * KC_ * KC_ + i * KC_ + j] * hw[((size_t)g * KC_ + j) * H_ + d];
  y[((size_t)g * KC_ + i) * H_ + d] = a + bias[d];
}

// per-channel sum/sumsq over 320 rows (one block per channel)
__global__ void k_stats320(const float* __restrict__ y, float* __restrict__ stats) {
  __shared__ float ss[64], qq[64];
  int c = blockIdx.x, t = threadIdx.x;
  float s = 0.0f, q = 0.0f;
  for (int r = t; r < G_ * KC_; r += 64) { float v = y[(size_t)r * H_ + c]; s += v; q += v * v; }
  ss[t] = s; qq[t] = q;
  __syncthreads();
  for (int o = 32; o > 0; o >>= 1) { if (t < o) { ss[t] += ss[t + o]; qq[t] += qq[t + o]; } __syncthreads(); }
  if (t == 0) { stats[c] = ss[0]; stats[16 + c] = qq[0]; }
}

// max over K clusters of concat(x21,x22,x23) -> [G,30]
__global__ void k_x2max(const float* __restrict__ a, const float* __restrict__ b,
                        const float* __restrict__ c3, float* __restrict__ out) {
  int g = blockIdx.x, t = threadIdx.x;
  if (t >= 30) return;
  const float* p = (t < 10) ? a : (t < 20) ? b : c3;
  int cc = t % 10;
  float m = -3.4e38f;
  for (int i = 0; i < KC_; ++i) m = fmaxf(m, p[(size_t)(g * KC_ + i) * H_ + cc]);
  out[g * 30 + t] = m;
}

// head: concat -> fc1(relu) -> fc2; block 0 also writes reg = el/N + ml
__global__ void k_head(const float* __restrict__ x1o, const float* __restrict__ x2o,
                       const float* __restrict__ w1, const float* __restrict__ b1,
                       const float* __restrict__ w2, const float* __restrict__ b2,
                       const float* __restrict__ scal, float* __restrict__ out) {
  __shared__ float cvec[60];
  __shared__ float hbuf[50];
  int g = blockIdx.x, t = threadIdx.x;
  if (t < 30) cvec[t] = x1o[g * 30 + t];
  else if (t < 60) cvec[t] = x2o[g * 30 + (t - 30)];
  __syncthreads();
  if (t < 50) {
    float a = b1[t];
    for (int j = 0; j < 60; ++j) a += cvec[j] * w1[j * 50 + t];
    hbuf[t] = fmaxf(a, 0.0f);
  }
  __syncthreads();
  if (t < 6) {
    float a = b2[t];
    for (int j = 0; j < 50; ++j) a += hbuf[j] * w2[j * 6 + t];
    out[g * 6 + t] = a;
  }
  if (g == 0 && t == 0) out[G_ * 6] = scal[2] / (float)N_ + scal[3];   // reg
}

// =====================================================================
extern "C" void kernel_launch(void* const* d_in, const int* in_sizes, int n_in,
                              void* d_out, int out_size, void* d_ws, size_t ws_size,
                              hipStream_t stream) {
  (void)in_sizes; (void)n_in; (void)out_size; (void)ws_size;
  const float* xn  = (const float*)d_in[0];
  const int*   ei  = (const int*)  d_in[1];
  const float* ea  = (const float*)d_in[2];
  const float* s1  = (const float*)d_in[3];
  const int* src = ei;
  const int* dst = ei + E_;
  // params flattened in dict order
  const float* w11   = (const float*)d_in[4];
  const float* b11   = (const float*)d_in[5];
  const float* bn11g = (const float*)d_in[6];
  const float* bn11b = (const float*)d_in[7];
  const float* w12   = (const float*)d_in[8];
  const float* b12   = (const float*)d_in[9];
  const float* bn12g = (const float*)d_in[10];
  const float* bn12b = (const float*)d_in[11];
  const float* w13   = (const float*)d_in[12];
  const float* b13   = (const float*)d_in[13];
  const float* bn13g = (const float*)d_in[14];
  const float* bn13b = (const float*)d_in[15];
  const float* w21   = (const float*)d_in[16];
  const float* b21   = (const float*)d_in[17];
  const float* bn21g = (const float*)d_in[18];
  const float* bn21b = (const float*)d_in[19];
  const float* w22   = (const float*)d_in[20];
  const float* b22   = (const float*)d_in[21];
  const float* bn22g = (const float*)d_in[22];
  const float* bn22b = (const float*)d_in[23];
  const float* w23   = (const float*)d_in[24];
  const float* b23   = (const float*)d_in[25];
  const float* bn23g = (const float*)d_in[26];
  const float* bn23b = (const float*)d_in[27];
  const float* poolw = (const float*)d_in[28];
  const float* poolb = (const float*)d_in[29];
  const float* fc1w  = (const float*)d_in[30];
  const float* fc1b  = (const float*)d_in[31];
  const float* fc2w  = (const float*)d_in[32];
  const float* fc2b  = (const float*)d_in[33];

  // ---- workspace bump allocation (floats), ~8.2 MB total ----
  size_t off = 0;
  auto A = [&](size_t n) { float* p = (float*)d_ws + off; off += n; return p; };
  float* deg   = A(N_);                 // deg_w (pre self-loop)
  float* dis   = A(N_);
  float* mm    = A((size_t)N_ * H_);    // x@W result (reused per layer)
  float* sc    = A((size_t)N_ * H_);    // scatter accumulator / pre-BN (reused)
  float* x11   = A((size_t)N_ * H_);
  float* x12   = A((size_t)N_ * H_);
  float* x13   = A((size_t)N_ * H_);
  float* sbuf  = A((size_t)N_ * KC_);   // pool logits -> softmax s
  float* x1o   = A(G_ * 30);
  float* p1x   = A(G_ * KC_ * H_);
  float* p1adj = A(G_ * KC_ * KC_);
  float* anorm = A(G_ * KC_ * KC_);
  float* mm2   = A(G_ * KC_ * H_);
  float* y2    = A(G_ * KC_ * H_);
  float* x21   = A(G_ * KC_ * H_);
  float* x22   = A(G_ * KC_ * H_);
  float* x23   = A(G_ * KC_ * H_);
  float* x2o   = A(G_ * 30);
  float* stats = A(6 * 32);             // 6 BN layers x (sum[16], sumsq[16])
  float* scal  = A(16);                 // [0]=2m [1]=q_edge [2]=el_sum [3]=ml
  float* dvec  = A(16);

  // one async memset re-zeroes all accumulators per launch (graph-capture safe)
  hipMemsetAsync(d_ws, 0, off * sizeof(float), stream);

  dim3 b256(256);

  // degree / normalization
  k_degree<<<E_ / 256, b256, 0, stream>>>(src, dst, ea, deg, scal + 0);
  k_dis<<<N_ / 256, b256, 0, stream>>>(deg, dis);

  // ---- node GCN stack (WMMA linear + scatter + self-term/stats + BN) ----
  auto gcn_tail = [&](const float* B, const float* bng, const float* bnb,
                      float* xout, float* st) {
    k_gcn_scatter<<<E_ / 256, b256, 0, stream>>>(src, dst, ea, dis, mm, sc);
    k_self_stats<<<N_ / 256, b256, 0, stream>>>(mm, dis, B, sc, st);
    k_bn_apply<<<(N_ * H_) / 256, b256, 0, stream>>>(sc, st, bng, bnb, xout, N_);
  };
  // layer 1: din=3
  k_wmma_linear<3, H_><<<(N_ / 16) / 8, b256, 0, stream>>>(xn, w11, mm, N_);
  gcn_tail(b11, bn11g, bn11b, x11, stats + 0);
  // layer 2: din=10
  k_zero<<<(N_ * H_) / 256, b256, 0, stream>>>(sc, N_ * H_);
  k_wmma_linear<H_, H_><<<(N_ / 16) / 8, b256, 0, stream>>>(x11, w12, mm, N_);
  gcn_tail(b12, bn12g, bn12b, x12, stats + 32);
  // layer 3: din=10
  k_zero<<<(N_ * H_) / 256, b256, 0, stream>>>(sc, N_ * H_);
  k_wmma_linear<H_, H_><<<(N_ / 16) / 8, b256, 0, stream>>>(x12, w13, mm, N_);
  gcn_tail(b13, bn13g, bn13b, x13, stats + 64);

  k_graph_max<<<G_, b256, 0, stream>>>(x11, x12, x13, x1o);

  // ---- assignment s = softmax(s1 @ pool_w + pool_b) ----
  k_wmma_linear<14, KC_><<<(N_ / 16) / 8, b256, 0, stream>>>(s1, poolw, sbuf, N_);
  k_softmax<<<N_ / 256, b256, 0, stream>>>(sbuf, poolb, N_);

  // ---- modularity + entropy regularizers ----
  k_mod_edge<<<E_ / 256, b256, 0, stream>>>(src, dst, ea, sbuf, scal + 1);
  k_dvec<<<N_ / 256, b256, 0, stream>>>(sbuf, deg, dvec);
  k_ml<<<1, 32, 0, stream>>>(scal, dvec);
  k_entropy<<<N_ / 256, b256, 0, stream>>>(sbuf, scal + 2);

  // ---- dense_diff_pool (edge-wise; never materialize dense adjacency) ----
  k_p1x_wmma<<<(G_ + 7) / 8, b256, 0, stream>>>(sbuf, x13, p1x);
  k_p1adj<<<G_, b256, 0, stream>>>(src, dst, ea, sbuf, p1adj);
  k_anorm<<<G_, 128, 0, stream>>>(p1adj, anorm);

  // ---- coarse GCN stack on [G,K,H] ----
  auto gcn2_layer = [&](const float* hin, const float* W, const float* B,
                        const float* bng, const float* bnb, float* xout, float* st) {
    int waves = (G_ * KC_) / 16;  // 20 tiles of 16 rows
    k_wmma_linear<H_, H_><<<(waves + 7) / 8, b256, 0, stream>>>(hin, W, mm2, G_ * KC_);
    k_gcn2<<<G_, 128, 0, stream>>>(anorm, mm2, B, y2);
    k_stats320<<<H_, 64, 0, stream>>>(y2, st);
    k_bn_apply<<<((G_ * KC_ * H_) + 255) / 256, b256, 0, stream>>>(y2, st, bng, bnb, xout, G_ * KC_);
  };
  gcn2_layer(p1x, w21, b21, bn21g, bn21b, x21, stats + 96);
  gcn2_layer(x21, w22, b22, bn22g, bn22b, x22, stats + 128);
  gcn2_layer(x22, w23, b23, bn23g, bn23b, x23, stats + 160);

  k_x2max<<<G_, 32, 0, stream>>>(x21, x22, x23, x2o);
  k_head<<<G_, 64, 0, stream>>>(x1o, x2o, fc1w, fc1b, fc2w, fc2b, scal, (float*)d_out);
}